// SMeta_38671885533528
// MI455X (gfx1250) — compile-verified
//
#include <hip/hip_runtime.h>

#define NN   2000   // nodes
#define NE   16000  // edges
#define BB   4      // batch
#define LL   12     // L1
#define IND  16     // in_dim
#define RF   32     // ref
#define HD   32     // hid
#define OD   16     // out
#define NCOL 2048   // 2*HD*HD

#define PAIRS 32    // (edge,batch) pairs per block
#define CCH   256   // meta_W columns per chunk
#define NCHK  (NCOL / CCH)   // 8 chunks
#define KCH   (CCH / HD)     // 8 st_e features per chunk

typedef __attribute__((ext_vector_type(2))) float v2f;
typedef __attribute__((ext_vector_type(8))) float v8f;

// ---------------------------------------------------------------------------
// K1: state[n,b,l,h] = x[n,b,l,:] @ W_in + b_in   (node-major layout in ws)
// ---------------------------------------------------------------------------
__global__ __launch_bounds__(256) void k_state(
    const float* __restrict__ sdi,   // [B,L1,N,IN]
    const float* __restrict__ W_in,  // [IN,HD]
    const float* __restrict__ b_in,  // [HD]
    float* __restrict__ state)       // [N,B,L1,HD]
{
  int idx = blockIdx.x * blockDim.x + threadIdx.x;
  if (idx >= NN * BB * LL * HD) return;
  int h   = idx & (HD - 1);
  int row = idx >> 5;          // ((n*B+b)*L1 + l)
  int l   = row % LL;
  int nb  = row / LL;
  int b   = nb & (BB - 1);
  int n   = nb >> 2;
  const float* xr = &sdi[((b * LL + l) * NN + n) * IND];
  float acc = b_in[h];
#pragma unroll
  for (int i = 0; i < IND; ++i) acc += xr[i] * W_in[i * HD + h];
  state[idx] = acc;
}

// ---------------------------------------------------------------------------
// K2: fused per-edge hypernetwork + attention + segment-softmax accumulation.
// One block = 32 (edge,batch) pairs (two 16-row WMMA M-tiles so every
// W_metaW B-fragment feeds two v_wmma_f32_16x16x4_f32 ops). B-fragments are
// staged in a register array (8 k-steps deep) so loads clause and pipeline.
// ---------------------------------------------------------------------------
__global__ __launch_bounds__(128) void k_edge(
    const float* __restrict__ longd,    // [B,N,RF]
    const int*   __restrict__ src,
    const int*   __restrict__ dst,
    const float* __restrict__ W_metaW,  // [2RF, NCOL]
    const float* __restrict__ b_metaW,  // [NCOL]
    const float* __restrict__ W_metab,  // [2RF, HD]
    const float* __restrict__ b_metab,  // [HD]
    const float* __restrict__ state,    // [N,B,L1,HD]
    float* __restrict__ denom,          // [N,B,L1,HD]
    float* __restrict__ num)            // [N,B,L1,HD]
{
  __shared__ float s_mk[PAIRS][64];         // mk_e tile (pairs x 2RF)      8 KB
  __shared__ float s_mb[PAIRS][HD];         // meta_b tile                  4 KB
  __shared__ float s_mw[PAIRS][CCH + 8];    // meta_W column-chunk         ~34 KB
  __shared__ int   s_sn[PAIRS], s_dn[PAIRS], s_bb[PAIRS];

  const int t    = threadIdx.x;
  const int wave = t >> 5;
  const int lane = t & 31;
  const int half = lane >> 4;
  const int lm   = lane & 15;
  const int p0   = blockIdx.x * PAIRS;

  if (t < PAIRS) {
    int p = p0 + t;
    int e = p >> 2;               // p / B
    s_sn[t] = src[e];
    s_dn[t] = dst[e];
    s_bb[t] = p & (BB - 1);
  }
  __syncthreads();

  // mk_e = concat(mk[src], mk[dst]) : 32*64 elements
  for (int i = t; i < PAIRS * 64; i += 128) {
    int p = i >> 6, j = i & 63;
    int node = (j < RF) ? s_sn[p] : s_dn[p];
    s_mk[p][j] = longd[(s_bb[p] * NN + node) * RF + (j & (RF - 1))];
  }
  __syncthreads();

  // meta_b = mk_e @ W_metab + b_metab (tiny, scalar)
  for (int i = t; i < PAIRS * HD; i += 128) {
    int p = i >> 5, h = i & (HD - 1);
    float acc = b_metab[h];
#pragma unroll 8
    for (int j = 0; j < 64; ++j) acc += s_mk[p][j] * W_metab[j * HD + h];
    s_mb[p][h] = acc;
  }

  // Preload GEMM-1 A fragments for both M-tiles (pairs 0..15 and 16..31).
  // Layout: VGPR0 -> K=4s+2*half, VGPR1 -> K=4s+2*half+1, M = lane&15.
  v2f aA0[16], aA1[16];
#pragma unroll
  for (int s = 0; s < 16; ++s) {
    int j0 = 4 * s + 2 * half;
    aA0[s].x = s_mk[lm][j0];
    aA0[s].y = s_mk[lm][j0 + 1];
    aA1[s].x = s_mk[16 + lm][j0];
    aA1[s].y = s_mk[16 + lm][j0 + 1];
  }

  // att accumulators: 8 pairs/wave x 2 N-tiles (h0 = 0,16), persist over chunks
  v8f attC[8][2];
#pragma unroll
  for (int d = 0; d < 8; ++d)
#pragma unroll
    for (int u = 0; u < 2; ++u)
      attC[d][u] = (v8f){0.f, 0.f, 0.f, 0.f, 0.f, 0.f, 0.f, 0.f};

  for (int ch = 0; ch < NCHK; ++ch) {
    const int c0 = ch * CCH;      // column base (col = k*HD + h)
    const int k0 = ch * KCH;      // covered k-range of st_e
    __syncthreads();

    // ---- GEMM 1: meta_W chunk = mk_e[32x64] @ W_metaW[64 x 256] (+bias) ----
    for (int q = 0; q < 4; ++q) {                 // 16 N-tiles / 4 waves
      int cloc = (wave * 4 + q) * 16;
      int col  = c0 + cloc + lm;
      v8f c0v = (v8f){0.f, 0.f, 0.f, 0.f, 0.f, 0.f, 0.f, 0.f};
      v8f c1v = (v8f){0.f, 0.f, 0.f, 0.f, 0.f, 0.f, 0.f, 0.f};
#pragma unroll
      for (int sh = 0; sh < 2; ++sh) {            // two 8-deep halves
        v2f bB[8];                                // staged B-fragments
#pragma unroll
        for (int s = 0; s < 8; ++s) {
          int j0 = 4 * (8 * sh + s) + 2 * half;
          bB[s].x = W_metaW[j0 * NCOL + col];
          bB[s].y = W_metaW[(j0 + 1) * NCOL + col];
        }
#pragma unroll
        for (int s = 0; s < 8; ++s) {
          c0v = __builtin_amdgcn_wmma_f32_16x16x4_f32(
                    false, aA0[8 * sh + s], false, bB[s], (short)0, c0v, false, false);
          c1v = __builtin_amdgcn_wmma_f32_16x16x4_f32(
                    false, aA1[8 * sh + s], false, bB[s], (short)0, c1v, false, false);
        }
      }
      float bias = b_metaW[col];
#pragma unroll
      for (int r = 0; r < 8; ++r) {               // D: VGPR r -> row r+8*half
        s_mw[r + 8 * half][cloc + lm]      = c0v[r] + bias;
        s_mw[16 + r + 8 * half][cloc + lm] = c1v[r] + bias;
      }
    }
    __syncthreads();

    // ---- GEMM 2: att += st_e[:,k0:k0+8] @ meta_W[k0:k0+8,:] per pair ----
#pragma unroll
    for (int d = 0; d < 8; ++d) {
      int pw = wave * 8 + d;
      int sn = s_sn[pw], dn = s_dn[pw], bb = s_bb[pw];
#pragma unroll
      for (int s = 0; s < KCH / 4; ++s) {         // 2 WMMA k-steps
        int kk0 = 4 * s + 2 * half;               // chunk-local k (even)
        v2f aS = (v2f){0.f, 0.f};
        int l = lm;                               // M = l, rows 12..15 pad 0
        if (l < LL) {
          int kg0 = k0 + kk0;                     // even -> kg0,kg0+1 same node
          int nd  = (kg0 < HD) ? sn : dn;
          aS = *(const v2f*)(state +
                 ((nd * BB + bb) * LL + l) * HD + (kg0 & (HD - 1)));
        }
#pragma unroll
        for (int u = 0; u < 2; ++u) {             // h0 = 16*u
          v2f bS;
          bS.x = s_mw[pw][(4 * s + 2 * half) * HD + 16 * u + lm];
          bS.y = s_mw[pw][(4 * s + 2 * half + 1) * HD + 16 * u + lm];
          attC[d][u] = __builtin_amdgcn_wmma_f32_16x16x4_f32(
                           false, aS, false, bS, (short)0, attC[d][u], false, false);
        }
      }
    }
  }

  // ---- Finalize: e = exp(att + meta_b); segment sums via float atomics ----
#pragma unroll
  for (int d = 0; d < 8; ++d) {
    int pw = wave * 8 + d;
    int sn = s_sn[pw], dn = s_dn[pw], bb = s_bb[pw];
#pragma unroll
    for (int u = 0; u < 2; ++u) {
      int h = 16 * u + lm;
      float mb = s_mb[pw][h];
#pragma unroll
      for (int r = 0; r < 8; ++r) {
        int l = r + 8 * half;                     // D row -> l
        if (l < LL) {
          float ee = __expf(attC[d][u][r] + mb);
          int di = ((dn * BB + bb) * LL + l) * HD + h;
          float sv = state[((sn * BB + bb) * LL + l) * HD + h];
          atomicAdd(&denom[di], ee);
          atomicAdd(&num[di], ee * sv);
        }
      }
    }
  }
}

// ---------------------------------------------------------------------------
// K3: pred = (num/denom') @ W_out + b_out, transposed to [B,L1,N,OD]
// ---------------------------------------------------------------------------
__global__ __launch_bounds__(256) void k_out(
    const float* __restrict__ num,
    const float* __restrict__ denom,
    const float* __restrict__ W_out,   // [HD,OD]
    const float* __restrict__ b_out,   // [OD]
    float* __restrict__ out)           // [B,L1,N,OD]
{
  int idx = blockIdx.x * blockDim.x + threadIdx.x;
  if (idx >= BB * LL * NN * OD) return;
  int o    = idx & (OD - 1);
  int rest = idx >> 4;                 // (b*L1+l)*N + n
  int n    = rest % NN;
  int bl   = rest / NN;
  int l    = bl % LL;
  int b    = bl / LL;
  int base = ((n * BB + b) * LL + l) * HD;
  float acc = b_out[o];
#pragma unroll 8
  for (int h = 0; h < HD; ++h) {
    float dd = denom[base + h];
    float ns = num[base + h] / ((dd == 0.f) ? 1.f : dd);
    acc += ns * W_out[h * OD + o];
  }
  out[idx] = acc;
}

// ---------------------------------------------------------------------------
extern "C" void kernel_launch(void* const* d_in, const int* in_sizes, int n_in,
                              void* d_out, int out_size, void* d_ws, size_t ws_size,
                              hipStream_t stream) {
  (void)in_sizes; (void)n_in; (void)out_size; (void)ws_size;
  const float* longd   = (const float*)d_in[0];
  const float* sdi     = (const float*)d_in[1];
  const int*   src     = (const int*)d_in[2];
  const int*   dst     = (const int*)d_in[3];
  const float* W_in    = (const float*)d_in[4];
  const float* b_in    = (const float*)d_in[5];
  const float* W_metaW = (const float*)d_in[6];
  const float* b_metaW = (const float*)d_in[7];
  const float* W_metab = (const float*)d_in[8];
  const float* b_metab = (const float*)d_in[9];
  const float* W_out   = (const float*)d_in[10];
  const float* b_out   = (const float*)d_in[11];
  float* out = (float*)d_out;

  float* ws = (float*)d_ws;
  const size_t SB = (size_t)NN * BB * LL * HD;  // 3,072,000 floats
  float* state = ws;
  float* denom = ws + SB;
  float* numb  = ws + 2 * SB;

  hipMemsetAsync(denom, 0, 2 * SB * sizeof(float), stream);

  k_state<<<(NN * BB * LL * HD + 255) / 256, 256, 0, stream>>>(sdi, W_in, b_in, state);
  k_edge<<<(NE * BB) / PAIRS, 128, 0, stream>>>(longd, src, dst, W_metaW, b_metaW,
                                                W_metab, b_metab, state, denom, numb);
  k_out<<<(BB * LL * NN * OD + 255) / 256, 256, 0, stream>>>(numb, denom, W_out, b_out, out);
}